// AnimeInterp_72043781423597
// MI455X (gfx1250) — compile-verified
//
#include <hip/hip_runtime.h>
#include <stdint.h>

// Softmax splatting (softsplat, strMode='soft') for MI455X / gfx1250.
// Strategy: scatter-add with native f32 L2 atomics; 133.7MB accumulator state
// (numerator in d_out, norm plane in d_ws) stays resident in the 192MB L2.
// Accumulator zero-fill uses the gfx1250 async LDS->global store path
// (GLOBAL_STORE_ASYNC_FROM_LDS_B128, tracked on ASYNCcnt).

#define C_CH 16
#define H_DIM 512
#define W_DIM 960

// ---------------------------------------------------------------------------
// Kernel: forward bilinear splat with exp(metric) weighting.
// One thread per source pixel; up to 4*17 native f32 atomics into L2.
// (Placed first in the file so the disasm snippet shows the atomic path.)
// ---------------------------------------------------------------------------
__global__ __launch_bounds__(256) void ss_splat_kernel(
    const float* __restrict__ ten_in,    // [B,16,H,W]
    const float* __restrict__ ten_flow,  // [B,2,H,W]
    const float* __restrict__ ten_metric,// [B,1,H,W]
    float* __restrict__ num,             // [B,16,H,W] accumulator (== d_out)
    float* __restrict__ nrm,             // [B,H,W]    accumulator (in d_ws)
    int B) {
    const int HW = H_DIM * W_DIM;
    const int N  = B * HW;
    const int idx = blockIdx.x * blockDim.x + threadIdx.x;
    if (idx >= N) return;

    const int b = idx / HW;
    const int p = idx - b * HW;
    const int y = p / W_DIM;
    const int x = p - y * W_DIM;

    // gfx1250 prefetch of the flow stream ahead of use.
    __builtin_prefetch(ten_flow + (size_t)(b * 2) * HW + p + 2048, 0, 1);

    const float fdx = ten_flow[(size_t)(b * 2 + 0) * HW + p];
    const float fdy = ten_flow[(size_t)(b * 2 + 1) * HW + p];
    const float fx  = (float)x + fdx;
    const float fy  = (float)y + fdy;
    const float m   = __expf(ten_metric[(size_t)b * HW + p]);

    const float x0f = floorf(fx);
    const float y0f = floorf(fy);
    const int x0 = (int)x0f, y0 = (int)y0f;
    const int x1 = x0 + 1,   y1 = y0 + 1;
    const float wx1 = fx - x0f, wx0 = 1.f - wx1;
    const float wy1 = fy - y0f, wy0 = 1.f - wy1;

    // Load this pixel's 16 channels (coalesced per plane), pre-scale by m.
    float v[C_CH];
    const float* inb = ten_in + (size_t)b * C_CH * HW + p;
#pragma unroll
    for (int c = 0; c < C_CH; ++c) v[c] = inb[(size_t)c * HW] * m;

    float* numb = num + (size_t)b * C_CH * HW;
    float* nrmb = nrm + (size_t)b * HW;

    const bool vx0 = (x0 >= 0) & (x0 < W_DIM);
    const bool vx1 = (x1 >= 0) & (x1 < W_DIM);
    const bool vy0 = (y0 >= 0) & (y0 < H_DIM);
    const bool vy1 = (y1 >= 0) & (y1 < H_DIM);

    auto splat1 = [&](int xi, int yi, float w) {
        const int off = yi * W_DIM + xi;
#pragma unroll
        for (int c = 0; c < C_CH; ++c)
            unsafeAtomicAdd(&numb[(size_t)c * HW + off], v[c] * w);   // global_atomic_add_f32
        unsafeAtomicAdd(&nrmb[off], m * w);
    };

    if (vx0 & vy0) splat1(x0, y0, wx0 * wy0);
    if (vx1 & vy0) splat1(x1, y0, wx1 * wy0);
    if (vx0 & vy1) splat1(x0, y1, wx0 * wy1);
    if (vx1 & vy1) splat1(x1, y1, wx1 * wy1);
}

// ---------------------------------------------------------------------------
// Kernel: zero accumulators via async LDS->global B128 stores (ASYNCcnt path).
// All lanes source a shared 16-byte zero tile in LDS; stores are fire-and-
// forget with no data VGPRs. Drained by s_wait_asynccnt + implicit wait-idle
// at s_endpgm, and the kernel boundary orders it before the splat kernel.
// ---------------------------------------------------------------------------
__global__ __launch_bounds__(256) void ss_zero_async_kernel(
    float4* __restrict__ a, int n4a, float4* __restrict__ b, int n4b) {
    __shared__ float4 ztile;
    if (threadIdx.x == 0) ztile = make_float4(0.f, 0.f, 0.f, 0.f);
    __syncthreads();

    // Low 32 bits of a generic LDS pointer are the LDS byte offset.
    const unsigned lds_addr = (unsigned)(uintptr_t)&ztile;
    const int tid    = blockIdx.x * blockDim.x + threadIdx.x;
    const int stride = gridDim.x * blockDim.x;

    for (int i = tid; i < n4a; i += stride) {
        const float4* gp = a + i;
        asm volatile("global_store_async_from_lds_b128 %0, %1, off"
                     :: "v"(gp), "v"(lds_addr) : "memory");
    }
    for (int i = tid; i < n4b; i += stride) {
        const float4* gp = b + i;
        asm volatile("global_store_async_from_lds_b128 %0, %1, off"
                     :: "v"(gp), "v"(lds_addr) : "memory");
    }
    asm volatile("s_wait_asynccnt 0x0" ::: "memory");
}

// ---------------------------------------------------------------------------
// Kernel: out[b,c,y,x] = num[b,c,y,x] / (nrm[b,y,x] + 1e-7)   (in place)
// ---------------------------------------------------------------------------
__global__ __launch_bounds__(256) void ss_norm_kernel(
    float* __restrict__ out, const float* __restrict__ nrm, int B) {
    const int HW = H_DIM * W_DIM;
    const int N  = B * HW;
    const int idx = blockIdx.x * blockDim.x + threadIdx.x;
    if (idx >= N) return;

    const int b = idx / HW;
    const int p = idx - b * HW;
    const float inv = 1.0f / (nrm[idx] + 1e-7f);

    float* ob = out + (size_t)b * C_CH * HW + p;
#pragma unroll
    for (int c = 0; c < C_CH; ++c) ob[(size_t)c * HW] *= inv;
}

// ---------------------------------------------------------------------------
extern "C" void kernel_launch(void* const* d_in, const int* in_sizes, int n_in,
                              void* d_out, int out_size, void* d_ws, size_t ws_size,
                              hipStream_t stream) {
    const float* ten_in     = (const float*)d_in[0];
    const float* ten_flow   = (const float*)d_in[1];
    const float* ten_metric = (const float*)d_in[2];
    float* out = (float*)d_out;     // doubles as numerator accumulator
    float* nrm = (float*)d_ws;      // norm-channel accumulator [B,H,W]

    const int HW = H_DIM * W_DIM;
    const int NP = in_sizes[2];          // B*H*W (metric element count)
    const int B  = NP / HW;

    const int n4a = (B * C_CH * HW) / 4; // d_out float4 count
    const int n4b = NP / 4;              // norm  float4 count
    ss_zero_async_kernel<<<2048, 256, 0, stream>>>((float4*)out, n4a, (float4*)nrm, n4b);

    const int blocks = (NP + 255) / 256;
    ss_splat_kernel<<<blocks, 256, 0, stream>>>(ten_in, ten_flow, ten_metric, out, nrm, B);
    ss_norm_kernel<<<blocks, 256, 0, stream>>>(out, nrm, B);
}